// C3DLoss_39281770889277
// MI455X (gfx1250) — compile-verified
//
#include <hip/hip_runtime.h>
#include <hip/hip_bf16.h>

// Problem constants (match reference: B=8, H=384, W=1280)
#define BB   8
#define HH   384
#define WW   1280
#define NIMG (HH * WW)          // 491520 points per batch image
#define NTOT (BB * NIMG)        // 3932160 total points (divisible by 256*1920... see launch)
#define GRID_ELEMS (BB * 3 * NIMG)   // 11796480 floats: output grid
#define MASK_ELEMS (NTOT)            // 3932160 floats: output mask

typedef __attribute__((ext_vector_type(2))) float v2f;
typedef __attribute__((ext_vector_type(8))) float v8f;

// ---------------------------------------------------------------------------
// Kernel 1: zero the scatter target (grid region of d_out).
// ---------------------------------------------------------------------------
__global__ void zero_grid_kernel(float* __restrict__ g, int n) {
    int i = blockIdx.x * blockDim.x + threadIdx.x;
    int stride = gridDim.x * blockDim.x;
    for (; i < n; i += stride) g[i] = 0.0f;
}

// ---------------------------------------------------------------------------
// Kernel 2: project points through K with V_WMMA_F32_16X16X4_F32 and
// scatter-add in-view points into the dense grid with global f32 atomics.
//
// WMMA mapping (per wave, per loop iteration = 32 points):
//   A (16x4) = K zero-padded:  lane m<16: a0=K[m][0], a1=K[m][1] (m<3 else 0)
//                              lane m>=16: a0=K[m-16][2], a1=0
//   B (4x16) = 16 points' xyz: lane n<16: b0=x_n, b1=y_n
//                              lane n>=16: b0=z_{n-16}, b1=0
//   D (16x16): VGPR r, lanes 0-15 hold uvb[r] of point n=lane.
// Two WMMAs cover the lower / upper 16 points of the wave; shuffles move
// the upper-half results back to lanes 16..31.
// ---------------------------------------------------------------------------
__global__ __launch_bounds__(256) void project_scatter_kernel(
    const float* __restrict__ xyz,
    const float* __restrict__ flow,
    const unsigned char* __restrict__ mask,   // JAX bool -> 1 byte/elem
    const float* __restrict__ Kmat,           // (B,3,3)
    float* __restrict__ grid)                 // (B,3,H,W), pre-zeroed
{
    const int lane = threadIdx.x & 31;
    const int m    = lane & 15;
    const int tid  = blockIdx.x * blockDim.x + threadIdx.x;
    const int stride = gridDim.x * blockDim.x;   // 491520 -> exactly 8 uniform iters

    for (int p = tid; p < NTOT; p += stride) {
        const int b = p / NIMG;               // wave-uniform (NIMG % 32 == 0)
        const int n = p - b * NIMG;
        const float* xb = xyz  + (size_t)b * 3 * NIMG;
        const float* fb = flow + (size_t)b * 3 * NIMG;

        // This lane's point (stop_gradient is a no-op in forward)
        const float x = xb[n]            + fb[n];
        const float y = xb[NIMG + n]     + fb[NIMG + n];
        const float z = xb[2 * NIMG + n] + fb[2 * NIMG + n];

        // --- A matrix: K padded to 16x4 ---
        const float* Kb = Kmat + b * 9;
        float a0, a1;
        if (lane < 16) { a0 = (m < 3) ? Kb[m * 3 + 0] : 0.0f;
                         a1 = (m < 3) ? Kb[m * 3 + 1] : 0.0f; }
        else           { a0 = (m < 3) ? Kb[m * 3 + 2] : 0.0f;
                         a1 = 0.0f; }
        v2f A; A.x = a0; A.y = a1;

        // --- WMMA #0: points held by lanes 0..15 ---
        const float z_lo = __shfl(z, m, 32);          // upper lanes need z of lane-16
        v2f B0; B0.x = (lane < 16) ? x : z_lo;
                B0.y = (lane < 16) ? y : 0.0f;
        v8f c = {};
        v8f d0 = __builtin_amdgcn_wmma_f32_16x16x4_f32(
                     false, A, false, B0, (short)0, c, false, false);

        // --- WMMA #1: points held by lanes 16..31 ---
        const float x_hi = __shfl(x, m + 16, 32);
        const float y_hi = __shfl(y, m + 16, 32);
        v2f B1; B1.x = (lane < 16) ? x_hi : z;        // upper lanes: own z (k=2 row)
                B1.y = (lane < 16) ? y_hi : 0.0f;
        v8f d1 = __builtin_amdgcn_wmma_f32_16x16x4_f32(
                     false, A, false, B1, (short)0, c, false, false);

        // --- Redistribute: lane l<16 -> d0 (own), lane l>=16 -> d1 from lane l-16
        const float u1 = __shfl(d1[0], m, 32);
        const float v1 = __shfl(d1[1], m, 32);
        const float w1 = __shfl(d1[2], m, 32);
        const float uu = (lane < 16) ? d0[0] : u1;
        const float vv = (lane < 16) ? d0[1] : v1;
        const float ww = (lane < 16) ? d0[2] : w1;

        // --- Perspective divide + banker's rounding (jnp.round == rint) ---
        const float denom = fmaxf(fabsf(ww), 1e-6f);
        const float uf = rintf(uu / denom);
        const float vf = rintf(vv / denom);

        const bool mk = (mask[p] != 0);
        const bool inview = (uf > 0.0f) & (uf < (float)WW) &
                            (vf > 0.0f) & (vf < (float)HH) &
                            (z > 0.1f) & mk;

        if (inview) {   // non-inview points scatter zeros in the reference: no-op
            const int ui = min(max((int)uf, 0), WW - 1);
            const int vi = min(max((int)vf, 0), HH - 1);
            float* g = grid + (size_t)b * 3 * NIMG + (size_t)vi * WW + ui;
            atomicAdd(g,            x);
            atomicAdd(g + NIMG,     y);
            atomicAdd(g + 2 * NIMG, z);
        }
    }
}

// ---------------------------------------------------------------------------
// Kernel 3: mask_flowed = any(grid != 0, axis=channel), written as 0/1 float.
// ---------------------------------------------------------------------------
__global__ void mask_kernel(const float* __restrict__ grid,
                            float* __restrict__ outm) {
    const int p = blockIdx.x * blockDim.x + threadIdx.x;
    if (p >= NTOT) return;
    const int b = p / NIMG;
    const int n = p - b * NIMG;
    const float* g = grid + (size_t)b * 3 * NIMG;
    const bool any = (g[n] != 0.0f) | (g[NIMG + n] != 0.0f) | (g[2 * NIMG + n] != 0.0f);
    outm[p] = any ? 1.0f : 0.0f;
}

// ---------------------------------------------------------------------------
// Launch: inputs in setup_inputs() order: xyz_grid, flow_grid, mask_grid, K.
// d_out = [grid (B*3*H*W f32) | mask_flowed (B*H*W f32)].
// ---------------------------------------------------------------------------
extern "C" void kernel_launch(void* const* d_in, const int* in_sizes, int n_in,
                              void* d_out, int out_size, void* d_ws, size_t ws_size,
                              hipStream_t stream) {
    const float*         xyz  = (const float*)d_in[0];
    const float*         flow = (const float*)d_in[1];
    const unsigned char* msk  = (const unsigned char*)d_in[2];
    const float*         Kmat = (const float*)d_in[3];

    float* grid = (float*)d_out;
    float* outm = grid + GRID_ELEMS;

    // 1) zero the accumulation target (scatter-add requires it every call)
    zero_grid_kernel<<<2048, 256, 0, stream>>>(grid, GRID_ELEMS);

    // 2) project + scatter. 1920*256 = 491520 threads; NTOT/491520 = 8 uniform
    //    loop iterations -> EXEC is all-ones at every WMMA.
    project_scatter_kernel<<<1920, 256, 0, stream>>>(xyz, flow, msk, Kmat, grid);

    // 3) channel-wise any() -> mask output
    mask_kernel<<<NTOT / 256, 256, 0, stream>>>(grid, outm);
}